// DimeNetPP_31129922961735
// MI455X (gfx1250) — compile-verified
//
#include <hip/hip_runtime.h>
#include <hip/hip_bf16.h>
#include <math.h>

// ---------------------------------------------------------------------------
// DimeNet-style pair distances + triplet angles for block-structured molecules.
//   B = 128 molecules, M = 48 atoms each, cutoff 1.5.
// Per center j: Gram matrix G[i,k] = (r_i - r_j) . (r_k - r_j) computed with
// V_WMMA_F32_16X16X4_F32 (K padded 3 -> 4), then elementwise
//   b2  = max(n2_i * n2_k - G^2, 0)           (|u x w|^2 identity)
//   ang = atan2(sqrt(b2), G)  masked by adjacency and i != k.
// One wave32 per (molecule, center): 6144 blocks x 32 threads, 9 WMMA / wave.
// ---------------------------------------------------------------------------

#define MOL_B 128
#define MOL_M 48
#define CUT2  2.25f   // 1.5^2

typedef __attribute__((ext_vector_type(2))) float v2f;
typedef __attribute__((ext_vector_type(8))) float v8f;

__global__ __launch_bounds__(32) void dimenet_dist_angle_kernel(
    const float* __restrict__ coords,   // [B*M, 3]
    float* __restrict__ dists,          // [B, M, M]
    float* __restrict__ angles)         // [B, M, M, M]
{
    const int bj   = blockIdx.x;        // 0 .. B*M-1  (center index)
    const int b    = bj / MOL_M;
    const int j    = bj - b * MOL_M;
    const int lane = threadIdx.x;       // 0..31 (wave32)

    __shared__ float sV[3][MOL_M];      // SoA displacement vectors r_i - r_j
    __shared__ float sn2[MOL_M];        // |r_i - r_j|^2
    __shared__ int   sadj[MOL_M];       // adjacency of edge (j, i)

    const float* mol = coords + (size_t)b * MOL_M * 3;
    const float jx = mol[j * 3 + 0];
    const float jy = mol[j * 3 + 1];
    const float jz = mol[j * 3 + 2];

    // Phase 1: displacements, squared norms, adjacency, dists row j.
    for (int i = lane; i < MOL_M; i += 32) {
        float x = mol[i * 3 + 0] - jx;
        float y = mol[i * 3 + 1] - jy;
        float z = mol[i * 3 + 2] - jz;
        sV[0][i] = x;
        sV[1][i] = y;
        sV[2][i] = z;
        float n2 = x * x + y * y + z * z;
        sn2[i] = n2;
        int adj = (i != j) && (n2 < CUT2);
        sadj[i] = adj;
        dists[(size_t)bj * MOL_M + i] = adj ? sqrtf(n2) : 0.0f;
    }
    __syncthreads();

    // Phase 2: build the three 16x4 f32 WMMA fragments (K = x,y,z,0).
    // ISA 16x4 f32 A layout: lanes 0-15 row m=L, VGPR0=K0, VGPR1=K1;
    // lanes 16-31 row m=L-16, VGPR0=K2, VGPR1=K3.  B mirrors A, and since
    // G = V * V^T both operands share the same fragments.
    const int  lo = lane & 15;
    const bool hi = lane >= 16;
    v2f f[3];
#pragma unroll
    for (int t = 0; t < 3; ++t) {
        int idx = t * 16 + lo;
        float x = sV[0][idx];
        float y = sV[1][idx];
        float z = sV[2][idx];
        f[t].x = hi ? z : x;
        f[t].y = hi ? 0.0f : y;
    }

    // Phase 3: 3x3 tiles of 16x16 Gram via WMMA, then angle math + stores.
    float* aout = angles + (size_t)bj * MOL_M * MOL_M;
    const int rowoff = hi ? 8 : 0;
#pragma unroll
    for (int ti = 0; ti < 3; ++ti) {
#pragma unroll
        for (int tk = 0; tk < 3; ++tk) {
            v8f c = {};
            // 8 args: (neg_a, A, neg_b, B, c_mod, C, reuse_a, reuse_b)
            c = __builtin_amdgcn_wmma_f32_16x16x4_f32(
                false, f[ti], false, f[tk], (short)0, c, false, false);

            const int   k    = tk * 16 + lo;
            const float n2k  = sn2[k];
            const int   adjk = sadj[k];
#pragma unroll
            for (int v = 0; v < 8; ++v) {
                const int   i    = ti * 16 + v + rowoff;
                const float a    = c[v];
                const float n2i  = sn2[i];
                float b2 = fmaxf(n2i * n2k - a * a, 0.0f);
                bool  tmask = (sadj[i] != 0) && (adjk != 0) && (i != k);
                float bv  = (b2 > 0.0f) ? sqrtf(b2) : 0.0f;
                float ang = atan2f(bv, a);
                aout[(size_t)i * MOL_M + k] = tmask ? ang : 0.0f;
            }
        }
    }
}

extern "C" void kernel_launch(void* const* d_in, const int* in_sizes, int n_in,
                              void* d_out, int out_size, void* d_ws, size_t ws_size,
                              hipStream_t stream)
{
    // inputs: d_in[0]=atomic_ns (int, unused), d_in[1]=coords (float [N,3]),
    //         d_in[2]=batch_node_vec (unused; structure implied by blocking)
    const float* coords = (const float*)d_in[1];

    float* dists  = (float*)d_out;                              // B*M*M floats
    float* angles = dists + (size_t)MOL_B * MOL_M * MOL_M;      // B*M*M*M floats

    dim3 grid(MOL_B * MOL_M);   // one wave32 per (molecule, center)
    dim3 block(32);
    dimenet_dist_angle_kernel<<<grid, block, 0, stream>>>(coords, dists, angles);
}